// BundleAdjustment_89077621719312
// MI455X (gfx1250) — compile-verified
//
#include <hip/hip_runtime.h>
#include <cstdint>

// Bundle-adjustment reprojection residual for MI455X (gfx1250, wave32).
//
//  - Pose table (2048 x 7 f32) staged into LDS via CDNA5 async-to-LDS copies,
//    scattered into 8-float (32 B) padded rows so each random pose gather in
//    the hot loop is exactly two aligned ds_load_b128 ops.
//  - Two edges per lane: all streaming traffic is b64/b128 vectorized
//    (int2 / float4), 8 uniform grid-stride iterations, no divergence.
//  - Roofline: ~40 B HBM per edge streaming -> ~168 MB -> ~7 us floor at
//    23.3 TB/s; gather tables (4 MB + 2 MB) are L2-resident (192 MB L2).

#define POSES_N   2048
#define E_TOTAL   4194304
#define BLOCK     256
#define GRID      1024

struct EdgeOut { float r2, th2; };

__device__ __forceinline__ EdgeOut edge_project(
    int pi, int ii, int si, int ti,
    const float2* __restrict__ patch_rt,
    const float*  __restrict__ elev,
    const float*  __restrict__ s_poses)   // padded: 8 floats per pose
{
    const float2 rt  = patch_rt[pi];
    const float  phi = elev[ii];

    float sph, cph, sth, cth;
    sincosf(phi,  &sph, &cph);
    sincosf(rt.y, &sth, &cth);

    const float rc = rt.x * cph;
    float px = rc * cth;
    float py = rc * sth;
    float pz = rt.x * sph;

    // source pose: two aligned 16B LDS loads
    const float4 A = *reinterpret_cast<const float4*>(&s_poses[8 * si]);      // t.xyz, qv.x
    const float4 B = *reinterpret_cast<const float4*>(&s_poses[8 * si + 4]);  // qv.y, qv.z, qw, pad
    const float qx = A.w, qy = B.x, qz = B.y, qw = B.z;

    float tx = 2.0f * (qy * pz - qz * py);
    float ty = 2.0f * (qz * px - qx * pz);
    float tz = 2.0f * (qx * py - qy * px);
    float gx = px + qw * tx + (qy * tz - qz * ty) + A.x;
    float gy = py + qw * ty + (qz * tx - qx * tz) + A.y;
    float gz = pz + qw * tz + (qx * ty - qy * tx) + A.z;

    // target pose (conjugate rotation)
    const float4 C = *reinterpret_cast<const float4*>(&s_poses[8 * ti]);
    const float4 D = *reinterpret_cast<const float4*>(&s_poses[8 * ti + 4]);
    const float dx = gx - C.x, dy = gy - C.y, dz = gz - C.z;
    const float ux = -C.w, uy = -D.x, uz = -D.y, uw = D.z;

    float t2x = 2.0f * (uy * dz - uz * dy);
    float t2y = 2.0f * (uz * dx - ux * dz);
    float t2z = 2.0f * (ux * dy - uy * dx);
    float lx = dx + uw * t2x + (uy * t2z - uz * t2y);
    float ly = dy + uw * t2y + (uz * t2x - ux * t2z);
    float lz = dz + uw * t2z + (ux * t2y - uy * t2x);

    EdgeOut o;
    o.r2  = sqrtf(lx * lx + ly * ly + lz * lz);
    o.th2 = atan2f(ly, lx);
    return o;
}

__global__ __launch_bounds__(BLOCK) void ba_reproj_kernel(
    const float*  __restrict__ poses,      // 2048*7
    const float2* __restrict__ patch_rt,   // 524288 x (r,theta)
    const float*  __restrict__ elev,       // 524288
    const float*  __restrict__ scale,      // 2
    const float4* __restrict__ baseline4,  // E/2 x (bl0,bl1,bl0',bl1')
    const float4* __restrict__ weights4,   // E/2
    const int2*   __restrict__ patch_idx2, // E/2
    const int2*   __restrict__ inv_pidx2,  // E/2
    const int2*   __restrict__ src_idx2,   // E/2
    const int2*   __restrict__ tgt_idx2,   // E/2
    float4*       __restrict__ out4)       // E/2
{
    __shared__ alignas(32) float s_poses[POSES_N * 8];   // 65,536 B, padded rows

    const int tid = threadIdx.x;

    // ---- CDNA5 async copy: poses -> LDS, dword scatter into padded rows ----
    {
        const uint32_t lds_base = (uint32_t)(uintptr_t)(&s_poses[0]);
        const uint64_t gbase    = (uint64_t)(uintptr_t)poses;
        for (int d = tid; d < POSES_N * 7; d += BLOCK) {        // 56 iters/lane
            uint32_t p  = (uint32_t)d / 7u;
            uint32_t rm = (uint32_t)d - 7u * p;
            uint32_t lds_off = lds_base + p * 32u + rm * 4u;
            uint64_t ga      = gbase + 4ull * (uint64_t)d;
            asm volatile("global_load_async_to_lds_b32 %0, %1, off"
                         :: "v"(lds_off), "v"(ga) : "memory");
        }
        asm volatile("s_wait_asynccnt 0" ::: "memory");
        __syncthreads();
    }

    const float s0 = scale[0];
    const float s1 = scale[1];

    const int NPAIR  = E_TOTAL / 2;                      // 2,097,152
    const int stride = BLOCK * GRID;                     // 262,144 -> 8 iters
    for (int p = blockIdx.x * BLOCK + tid; p < NPAIR; p += stride) {
        const int pn = p + stride;
        if (pn < NPAIR) {                                // global_prefetch_b8
            __builtin_prefetch(&baseline4[pn], 0, 1);
            __builtin_prefetch(&weights4[pn],  0, 1);
        }

        const int2 pi = patch_idx2[p];
        const int2 ii = inv_pidx2[p];
        const int2 si = src_idx2[p];
        const int2 ti = tgt_idx2[p];

        const EdgeOut e0 = edge_project(pi.x, ii.x, si.x, ti.x, patch_rt, elev, s_poses);
        const EdgeOut e1 = edge_project(pi.y, ii.y, si.y, ti.y, patch_rt, elev, s_poses);

        const float4 bl = baseline4[p];
        const float4 w  = weights4[p];
        float4 o;
        o.x = (e0.r2  * s0 - bl.x) * sqrtf(w.x + 1e-8f);
        o.y = (e0.th2 * s1 - bl.y) * sqrtf(w.y + 1e-8f);
        o.z = (e1.r2  * s0 - bl.z) * sqrtf(w.z + 1e-8f);
        o.w = (e1.th2 * s1 - bl.w) * sqrtf(w.w + 1e-8f);
        out4[p] = o;
    }
}

extern "C" void kernel_launch(void* const* d_in, const int* in_sizes, int n_in,
                              void* d_out, int out_size, void* d_ws, size_t ws_size,
                              hipStream_t stream) {
    (void)in_sizes; (void)n_in; (void)out_size; (void)d_ws; (void)ws_size;
    ba_reproj_kernel<<<GRID, BLOCK, 0, stream>>>(
        (const float*)d_in[0],
        (const float2*)d_in[1],
        (const float*)d_in[2],
        (const float*)d_in[3],
        (const float4*)d_in[4],
        (const float4*)d_in[5],
        (const int2*)d_in[6],
        (const int2*)d_in[7],
        (const int2*)d_in[8],
        (const int2*)d_in[9],
        (float4*)d_out);
}